// TVDeconvolution2D_12549894439473
// MI455X (gfx1250) — compile-verified
//
#include <hip/hip_runtime.h>
#include <math.h>

// TV-Chambolle (skimage) fixed-iteration denoise on (B,T,F)=(8,2048,128) fp32.
// Memory/latency-bound stencil: working set (~32MB) is L2-resident on MI455X
// (192MB L2), so we optimize for launch count + L2 traffic, not FLOPs.
// One fused kernel per Chambolle step: a workgroup owns ALL 8 b-planes of an
// 8-row t-slab (b-halo internal, only 1 t-halo row), computes out=x+div(p)
// into LDS and updates p in place. The x tile is staged with the CDNA5 async
// global->LDS path (global_load_async_to_lds_b128 / s_wait_asynccnt).

#define B_DIM  8
#define T_DIM  2048
#define F_DIM  128
#define PLANE  (T_DIM * F_DIM)        // 262144
#define N_ELEM (B_DIM * PLANE)        // 2097152
#define TAU    (1.0f / 6.0f)          // 1/(2*ndim)
#define N_ITERS 29                    // MAX_ITERS - 1 scan steps

#define TILE_T   8                    // t-rows updated per workgroup
#define ROWS_PT  (TILE_T + 1)         // +1 t-halo row
#define TROWS    (B_DIM * ROWS_PT)    // 72 staged rows (all 8 planes)
#define TILE_N   (TROWS * F_DIM)      // 9216 floats per LDS tile (36,864 B)
#define UPD_N    (B_DIM * TILE_T * F_DIM)  // 8192 updated elements per WG

// ---- CDNA5 async global->LDS staging (guarded; falls back to sync copy) ----
#if defined(__has_builtin)
#  if __has_builtin(__builtin_amdgcn_global_load_async_to_lds_b128) && \
      __has_builtin(__builtin_amdgcn_s_wait_asynccnt)
#    define USE_ASYNC_LDS 1
#  endif
#endif

// Builtin pointer params are int4-typed:
//   (int4 addrspace(1)* gsrc, int4 addrspace(3)* ldst, imm offset, imm cpol)
typedef int v4i __attribute__((ext_vector_type(4)));
typedef __attribute__((address_space(1))) v4i gv4i;   // global
typedef __attribute__((address_space(3))) v4i lv4i;   // LDS

// --------------------------------------------------------------------------
__global__ __launch_bounds__(256) void zero_p_kernel(float* __restrict__ p) {
    int i = blockIdx.x * 256 + threadIdx.x;
    if (i < 3 * N_ELEM) p[i] = 0.0f;
}

// One full Chambolle step:
//   out = x + div(p);  g = grad(out);  p = (p - TAU*g) / (1 + (TAU/w)|g|)
__global__ __launch_bounds__(256) void step_kernel(
    const float* __restrict__ x, float* __restrict__ p,
    const float* __restrict__ lambdas) {
    __shared__ __align__(16) float sOut[TILE_N];  // out tile: 8 planes x 9 rows x 128
    __shared__ __align__(16) float sX[TILE_N];    // x tile, async-staged

    const int tid = threadIdx.x;
    const int t0  = blockIdx.x * TILE_T;          // 256 t-slabs

    float* p0 = p;
    float* p1 = p + N_ELEM;
    float* p2 = p + 2 * N_ELEM;

    // ---- Phase 0: async-stage x tile (72 rows x 512B = 2304 float4s) ------
    for (int v = tid; v < TROWS * (F_DIM / 4); v += 256) {   // 9 iterations
        int row = v >> 5;                  // 0..71
        int b   = row / ROWS_PT;           // 0..7
        int rr  = row - b * ROWS_PT;       // 0..8
        int t   = t0 + rr; if (t > T_DIM - 1) t = T_DIM - 1;   // clamp (guarded use)
        int c   = (v & 31) * 4;
        int gidx = (b * T_DIM + t) * F_DIM + c;
#if defined(USE_ASYNC_LDS)
        __builtin_amdgcn_global_load_async_to_lds_b128(
            (gv4i*)(x + gidx), (lv4i*)(&sX[v * 4]), 0, 0);
#else
        *(float4*)(&sX[v * 4]) = *(const float4*)(x + gidx);
#endif
    }
#if defined(USE_ASYNC_LDS)
    __builtin_amdgcn_s_wait_asynccnt(0);   // this wave's async copies landed
#endif
    __syncthreads();                       // all waves' copies visible

    // ---- Phase 1: out = x + div(p) into LDS (9216 elems, 36/thread) -------
    // div(p)[b,t,f] = (b>0?p0[b-1]:0)-p0 + (t>0?p1[t-1]:0)-p1 + (f>0?p2[f-1]:0)-p2
    for (int v = tid; v < TILE_N; v += 256) {                // 36 iterations
        int b   = v / (ROWS_PT * F_DIM);
        int rem = v - b * (ROWS_PT * F_DIM);
        int rr  = rem >> 7;                // 0..8
        int f   = rem & (F_DIM - 1);
        int t   = t0 + rr; if (t > T_DIM - 1) t = T_DIM - 1;   // clamp (guarded use)
        int i   = (b * T_DIM + t) * F_DIM + f;
        float d = -p0[i] - p1[i] - p2[i];
        if (b) d += p0[i - PLANE];
        if (t) d += p1[i - F_DIM];
        if (f) d += p2[i - 1];
        sOut[v] = sX[v] + d;
    }
    __syncthreads();

    // ---- Phase 2: gradients from LDS, update p (8192 elems, 32/thread) ----
    const float w   = lambdas[0];
    const float fac = TAU / w;
    for (int k = 0; k < UPD_N / 256; ++k) {                  // 32 iterations
        int e  = tid + k * 256;
        int b  = e >> 10;                  // /(TILE_T*F_DIM)
        int rr = (e >> 7) & (TILE_T - 1);  // 0..7
        int f  = e & (F_DIM - 1);
        int t  = t0 + rr;                  // always < T_DIM
        int so = (b * ROWS_PT + rr) * F_DIM + f;
        float uc = sOut[so];
        float g0 = (b < B_DIM - 1) ? (sOut[so + ROWS_PT * F_DIM] - uc) : 0.0f;
        float g1 = (t < T_DIM - 1) ? (sOut[so + F_DIM]           - uc) : 0.0f;
        float g2 = (f < F_DIM - 1) ? (sOut[so + 1]               - uc) : 0.0f;
        float norm = sqrtf(g0 * g0 + g1 * g1 + g2 * g2) * fac + 1.0f;
        float inv  = 1.0f / norm;
        int i = (b * T_DIM + t) * F_DIM + f;
        p0[i] = (p0[i] - TAU * g0) * inv;
        p1[i] = (p1[i] - TAU * g1) * inv;
        p2[i] = (p2[i] - TAU * g2) * inv;
    }
}

// Final pass: out = x + div(p) + bias[f]   (writes d_out directly)
__global__ __launch_bounds__(256) void final_out_kernel(
    const float* __restrict__ x, const float* __restrict__ p,
    const float* __restrict__ bias, float* __restrict__ out) {
    int i = blockIdx.x * 256 + threadIdx.x;
    const float* p0 = p;
    const float* p1 = p + N_ELEM;
    const float* p2 = p + 2 * N_ELEM;
    int f  = i & (F_DIM - 1);
    int t  = (i >> 7) & (T_DIM - 1);
    int bb = i >> 18;
    float d = -p0[i] - p1[i] - p2[i];
    if (bb) d += p0[i - PLANE];
    if (t)  d += p1[i - F_DIM];
    if (f)  d += p2[i - 1];
    out[i] = x[i] + d + bias[f];
}

// --------------------------------------------------------------------------
extern "C" void kernel_launch(void* const* d_in, const int* in_sizes, int n_in,
                              void* d_out, int out_size, void* d_ws, size_t ws_size,
                              hipStream_t stream) {
    (void)in_sizes; (void)n_in; (void)out_size; (void)ws_size;
    const float* x       = (const float*)d_in[0];   // (8,2048,128) fp32
    const float* lambdas = (const float*)d_in[1];   // (1,1) fp32, scalar weight
    const float* bias    = (const float*)d_in[2];   // (1,1,128) fp32

    float* outp = (float*)d_out;
    float* p    = (float*)d_ws;          // 3*N floats = 24 MB scratch

    zero_p_kernel<<<(3 * N_ELEM + 255) / 256, 256, 0, stream>>>(p);

    for (int it = 0; it < N_ITERS; ++it) {
        step_kernel<<<T_DIM / TILE_T, 256, 0, stream>>>(x, p, lambdas);
    }
    final_out_kernel<<<N_ELEM / 256, 256, 0, stream>>>(x, p, bias, outp);
}